// PeakLoss_53145925320952
// MI455X (gfx1250) — compile-verified
//
#include <hip/hip_runtime.h>
#include <stdint.h>

// ---------------------------------------------------------------------------
// PeakLoss on MI455X (gfx1250).
// HBM-bound: 64 MiB input @ 23.3 TB/s ~= 2.9 us floor. Single fused pass:
//   s = avg3x3(x)/9 (separable, zero pad) staged in LDS,
//   maxpool17 separable (row max then col max, -inf pad) -> sum,
//   avgpool17 mean folded into an analytic weighted sum of s,
//   per-wave reduction via v_wmma_f32_16x16x4_f32, deterministic finalize.
// Data movement uses global_load_async_to_lds_b128 (ASYNCcnt) when available.
// ---------------------------------------------------------------------------

#define WIMG 512
#define HIMG 512
#define BH 32                 // output rows per workgroup
#define ROWS_A 50             // BH + 18 sali rows (3x3 halo + 17x17 halo)
#define ROWS_S 48             // BH + 16 s rows
#define LSTRIDE 516           // floats per LDS row (+4 pad -> bank stagger)
#define NBLK_PER_IMG (HIMG / BH)
#define NIMG 64               // 2 tensors x 32 batch
#define NWG (NIMG * NBLK_PER_IMG)

typedef float v2f  __attribute__((ext_vector_type(2)));
typedef float v8f  __attribute__((ext_vector_type(8)));
typedef float f32x4 __attribute__((ext_vector_type(4)));

// Builtin prototype (from hipcc diagnostic): param0 = int4 addrspace(1)*,
// param1 = LDS side, then imm offset, imm cpol.
typedef int v4i __attribute__((vector_size(16)));
typedef v4i __attribute__((address_space(1)))* global_v4i_ptr;
typedef v4i __attribute__((address_space(3)))* lds_v4i_ptr;

#if defined(__has_builtin)
#if __has_builtin(__builtin_amdgcn_global_load_async_to_lds_b128)
#define HAVE_ASYNC_LDS 1
#endif
#endif
#ifndef HAVE_ASYNC_LDS
#define HAVE_ASYNC_LDS 0
#endif

// Number of 17-windows (stride 1, pad 8, clipped to [0,511]) covering q.
__device__ __forceinline__ float edge_cnt(int q) {
    int c = 17;
    if (q < 8) c -= (8 - q);
    if (q > HIMG - 9) c -= (q - (HIMG - 9));
    return (float)c;
}

__global__ void __launch_bounds__(256)
peak_block_kernel(const float* __restrict__ s0, const float* __restrict__ s1,
                  float* __restrict__ partials) {
    extern __shared__ float smem[];
    float* Abuf = smem;                       // ROWS_A x LSTRIDE
    float* Bbuf = smem + ROWS_A * LSTRIDE;    // ROWS_A x LSTRIDE
    __shared__ float wred[16];

    const int tid  = threadIdx.x;
    const int wg   = blockIdx.x;
    const int img  = wg >> 4;                 // / NBLK_PER_IMG
    const int rblk = wg & (NBLK_PER_IMG - 1);
    const int r0   = rblk * BH;
    const float* src = (img < 32) ? (s0 + (size_t)img * WIMG * HIMG)
                                  : (s1 + (size_t)(img - 32) * WIMG * HIMG);

    const float NEG_INF = -__builtin_inff();

    // ---- Stage 1: stage sali rows [r0-9, r0+BH+9) into Abuf (zeros outside).
    // 50 rows x 128 float4 = 6400 xfers -> 25 per thread.
    for (int k = 0; k < 25; ++k) {
        int idx  = tid + (k << 8);
        int row  = idx >> 7;                  // 128 float4 per row
        int c4   = idx & 127;
        int gr   = r0 - 9 + row;
        float* dst = Abuf + row * LSTRIDE + (c4 << 2);   // 16B aligned
        if (gr >= 0 && gr < HIMG) {
            const float* gsrc = src + (size_t)gr * WIMG + (c4 << 2);
#if HAVE_ASYNC_LDS
            __builtin_amdgcn_global_load_async_to_lds_b128(
                (global_v4i_ptr)(uintptr_t)(const void*)gsrc,
                (lds_v4i_ptr)(uint32_t)(uintptr_t)(void*)dst,
                0, 0);
#else
            *(f32x4*)dst = *(const f32x4*)gsrc;
#endif
        } else {
            f32x4 z = {0.f, 0.f, 0.f, 0.f};
            *(f32x4*)dst = z;
        }
    }
#if HAVE_ASYNC_LDS
#if defined(__has_builtin) && __has_builtin(__builtin_amdgcn_s_wait_asynccnt)
    __builtin_amdgcn_s_wait_asynccnt(0);
#else
    asm volatile("s_wait_asynccnt 0" ::: "memory");
#endif
#endif
    __syncthreads();

    // ---- Stage 2: horizontal 3-sum (zero pad) Abuf -> Bbuf. 50*512/256 = 100.
    for (int k = 0; k < 100; ++k) {
        int idx = tid + (k << 8);
        int row = idx >> 9;
        int x   = idx & 511;
        const float* ar = Abuf + row * LSTRIDE;
        float v = ar[x];
        if (x > 0)    v += ar[x - 1];
        if (x < 511)  v += ar[x + 1];
        Bbuf[row * LSTRIDE + x] = v;
    }
    __syncthreads();

    // ---- Stage 3: vertical 3-sum /9 -> s into Abuf rows 0..47.
    //      Rows outside image -> -inf (clipped max windows).
    //      Also accumulate the analytic avgpool17 weighted sum for owned rows.
    float wacc = 0.f;
    for (int k = 0; k < 96; ++k) {            // 48*512/256
        int idx = tid + (k << 8);
        int i   = idx >> 9;
        int x   = idx & 511;
        int sr  = r0 - 8 + i;
        float sv;
        if (sr >= 0 && sr < HIMG) {
            sv = (Bbuf[i * LSTRIDE + x] + Bbuf[(i + 1) * LSTRIDE + x] +
                  Bbuf[(i + 2) * LSTRIDE + x]) * (1.0f / 9.0f);
            if (sr >= r0 && sr < r0 + BH)
                wacc += sv * edge_cnt(x) * edge_cnt(sr);
        } else {
            sv = NEG_INF;
        }
        Abuf[i * LSTRIDE + x] = sv;
    }
    __syncthreads();

    // ---- Stage 4: horizontal 17-max (-inf pad) Abuf -> Bbuf, 8 px/thread.
    {
        const int chunk = tid & 63;
        const int rsub  = tid >> 6;           // 0..3
        const int x0    = chunk << 3;
        for (int rb = 0; rb < ROWS_S; rb += 4) {
            int i = rb + rsub;
            const float* ar = Abuf + i * LSTRIDE;
            float t[24];
#pragma unroll
            for (int k = 0; k < 24; ++k) {
                int x = x0 - 8 + k;
                t[k] = (x >= 0 && x < WIMG) ? ar[x] : NEG_INF;
            }
            float* br = Bbuf + i * LSTRIDE + x0;
#pragma unroll
            for (int ii = 0; ii < 8; ++ii) {
                float m = t[ii];
#pragma unroll
                for (int k = 1; k < 17; ++k) m = fmaxf(m, t[ii + k]);
                br[ii] = m;
            }
        }
    }
    __syncthreads();

    // ---- Stage 5: vertical 17-max + accumulate. Thread owns a column,
    //      8 output rows per register tile (rows j..j+16 within 0..47).
    float acc = 0.f;
    for (int h = 0; h < 2; ++h) {
        int x = tid + (h << 8);
        for (int j0 = 0; j0 < BH; j0 += 8) {
            float t[24];
#pragma unroll
            for (int k = 0; k < 24; ++k) t[k] = Bbuf[(j0 + k) * LSTRIDE + x];
#pragma unroll
            for (int ii = 0; ii < 8; ++ii) {
                float m = t[ii];
#pragma unroll
                for (int k = 1; k < 17; ++k) m = fmaxf(m, t[ii + k]);
                acc += m;
            }
        }
    }

    // ---- Per-wave reduction with one exact-f32 WMMA (uniform across block,
    //      EXEC all-ones as required). A(16x4): lane L, vgpr v -> M=L%16,
    //      K=2*(L/16)+v, so a={acc,wacc} gives A[M][0/2]=acc, A[M][1/3]=wacc.
    //      B(4x16) selector: B[0][0]=B[1][0]=1 (lane 0), B[2][1]=B[3][1]=1
    //      (lane 17) -> D[M][0]=acc_M+wacc_M, D[M][1]=acc_{M+16}+wacc_{M+16}.
    {
        const int lane = tid & 31;
        v2f a; a.x = acc; a.y = wacc;
        float bsel = (lane == 0 || lane == 17) ? 1.0f : 0.0f;
        v2f b; b.x = bsel; b.y = bsel;
        v8f c = {0.f, 0.f, 0.f, 0.f, 0.f, 0.f, 0.f, 0.f};
        c = __builtin_amdgcn_wmma_f32_16x16x4_f32(
                /*neg_a=*/false, a, /*neg_b=*/false, b,
                /*c_mod=*/(short)0, c, /*reuse_a=*/false, /*reuse_b=*/false);
        // Sum the 8 row-VGPRs, then fold the two lane-halves (rows 0-7 / 8-15).
        float e = c[0] + c[1] + c[2] + c[3] + c[4] + c[5] + c[6] + c[7];
        e += __shfl_xor(e, 16, 32);
        const int wv = tid >> 5;
        if (lane < 2) wred[wv * 2 + lane] = e;  // lane0: Σ(acc+wacc) lanes 0-15, lane1: lanes 16-31
    }
    __syncthreads();
    if (tid == 0) {
        float at = 0.f, wt = 0.f;
        for (int w = 0; w < 8; ++w) { at += wred[2 * w]; wt += wred[2 * w + 1]; }
        partials[2 * blockIdx.x]     = at;
        partials[2 * blockIdx.x + 1] = wt;
    }
}

// Deterministic fixed-order finalize: one block, f64 tree reduction.
__global__ void __launch_bounds__(256)
finalize_kernel(const float* __restrict__ partials, float* __restrict__ out) {
    __shared__ double sa[256], sw[256];
    const int t = threadIdx.x;
    double a = 0.0, w = 0.0;
    for (int i = t; i < NWG; i += 256) {
        a += (double)partials[2 * i];
        w += (double)partials[2 * i + 1];
    }
    sa[t] = a; sw[t] = w;
    __syncthreads();
    for (int s = 128; s > 0; s >>= 1) {
        if (t < s) { sa[t] += sa[t + s]; sw[t] += sw[t + s]; }
        __syncthreads();
    }
    if (t == 0) {
        const double E = 32.0 * 512.0 * 512.0;        // elements per tensor
        double diff = (sa[0] - sw[0] / 289.0) / (2.0 * E);
        out[0] = (float)(1.0 - diff);
    }
}

extern "C" void kernel_launch(void* const* d_in, const int* in_sizes, int n_in,
                              void* d_out, int out_size, void* d_ws, size_t ws_size,
                              hipStream_t stream) {
    (void)in_sizes; (void)n_in; (void)out_size; (void)ws_size;
    const float* s0 = (const float*)d_in[0];
    const float* s1 = (const float*)d_in[1];
    float* partials = (float*)d_ws;           // 2 * NWG floats (8 KB)
    float* out      = (float*)d_out;

    const size_t shmem = (size_t)2 * ROWS_A * LSTRIDE * sizeof(float); // ~202 KB
    (void)hipFuncSetAttribute((const void*)peak_block_kernel,
                              hipFuncAttributeMaxDynamicSharedMemorySize,
                              (int)shmem);

    peak_block_kernel<<<dim3(NWG), dim3(256), shmem, stream>>>(s0, s1, partials);
    finalize_kernel<<<dim3(1), dim3(256), 0, stream>>>(partials, out);
}